// InvNUConv2d_85323820302551
// MI455X (gfx1250) — compile-verified
//
#include <hip/hip_runtime.h>

// ---------------------------------------------------------------------------
// Label-grouped 1x1 conv (InvNUConv2d) for MI455X / gfx1250, wave32.
//   y[b,o,h,w] = sum_c weight[labels[b,h,w], o, c] * x[b,c,h,w] + bias[label]
//
// HBM-bound: 34.6 MB mandatory traffic -> ~1.5 us @ 23.3 TB/s. Strategy:
//  * weights (25.6 KB) staged in LDS once per block via TDM tensor_load_to_lds
//    (6-arg toolchain variant) -- tracked with TENSORcnt
//  * each wave owns 4 tiles of 16 consecutive pixels; x tile loaded once into
//    B-matrix layout registers with compile-time IOFFSETs off one base pointer
//  * ballot-scan over distinct labels in the tile (~10-12 expected of 25);
//    per label: A = W[l] from LDS (ds_load_b64), B = x masked by label match,
//    4x V_WMMA_F32_16X16X4_F32 accumulate (exact f32, K=16 in chunks of 4)
//  * label loop is scalar-uniform so EXEC is all-ones at every WMMA (ISA req)
// ---------------------------------------------------------------------------

typedef __attribute__((ext_vector_type(2))) float        v2f;
typedef __attribute__((ext_vector_type(8))) float        v8f;
typedef __attribute__((ext_vector_type(4))) unsigned int u32x4;
typedef __attribute__((ext_vector_type(4))) int          i32x4;
typedef __attribute__((ext_vector_type(8))) int          i32x8;

#define BB 4
#define CC 16
#define HH 256
#define WW 256
#define LL 25
#define HWSZ (HH * WW)                 // 65536 pixels per batch image
#define NPIX (BB * HWSZ)               // 262144 total pixels
#define WTOTAL (LL * CC * CC)          // 6400 weight floats (25.6 KB)
#define THREADS 256
#define WAVES_PER_BLOCK (THREADS / 32)
#define TILES_PER_WAVE 4               // 64 consecutive pixels per wave
#define PIX_PER_BLOCK (WAVES_PER_BLOCK * TILES_PER_WAVE * 16)  // 512
#define NBLOCKS (NPIX / PIX_PER_BLOCK)                          // 512

__global__ __launch_bounds__(THREADS)
void invnuconv_wmma_kernel(const float* __restrict__ x,
                           const int* __restrict__ labels,
                           const float* __restrict__ weight,
                           const float* __restrict__ bias,
                           float* __restrict__ y) {
  __shared__ float lw[WTOTAL];         // only LDS alloc -> byte offset 0

  const int tid  = threadIdx.x;
  const int lane = tid & 31;
  const int wave = tid >> 5;
  const int p    = lane & 15;          // pixel column / A-matrix row
  const int half = lane >> 4;          // K-half selector (ISA f32 A/B layouts)

  // ---- Stage all 25 16x16 weight matrices into LDS ------------------------
#if __has_builtin(__builtin_amdgcn_tensor_load_to_lds)
  if (wave == 0) {                     // TDM op is scalar; one wave issues it
    const unsigned long long ga = (unsigned long long)(uintptr_t)weight;
    // D# group0: count=1 | lds_addr=0 | global_addr[56:0] | type=2
    u32x4 g0 = {1u, 0u, (unsigned int)ga,
                (unsigned int)((ga >> 32) & 0x1FFFFFFu) | (2u << 30)};
    // D# group1: data_size=4B; tensor_dim0=tile_dim0=6400; dim1=tile_dim1=1;
    //            tensor_dim0_stride=6400; everything else 0
    i32x8 g1 = {(int)(2u << 16),                 // wg_mask=0, data_size=2 (4B)
                (int)((WTOTAL & 0xFFFF) << 16),  // tensor_dim0 lo16 @ bits63:48
                (int)((WTOTAL >> 16) | (1 << 16)),  // dim0 hi16, tensor_dim1=1
                (int)((WTOTAL & 0xFFFF) << 16),  // tile_dim0 @ bits127:112
                1,                               // tile_dim1=1, tile_dim2=0
                WTOTAL,                          // tensor_dim0_stride lo32
                0, 0};
    i32x4 gz4 = {0, 0, 0, 0};                    // groups 2/3 unused (2-D)
    i32x8 gz8 = {0, 0, 0, 0, 0, 0, 0, 0};
    __builtin_amdgcn_tensor_load_to_lds(g0, g1, gz4, gz4, gz8, 0);
    __builtin_amdgcn_s_wait_tensorcnt(0);
  }
  __syncthreads();
#else
  for (int i = tid; i < WTOTAL; i += THREADS) lw[i] = weight[i];
  __syncthreads();
#endif

  // ---- Per-wave bases: fold lane terms once, all loop offsets constant ----
  const int wave_gid = blockIdx.x * WAVES_PER_BLOCK + wave;
  const int qwave    = wave_gid * (TILES_PER_WAVE * 16);  // flat [B,H,W]
  const int b        = qwave >> 16;                       // uniform per wave
  const int pixw     = qwave & (HWSZ - 1);

  const float* xph  = x + b * (CC * HWSZ) + 2 * half * HWSZ + pixw + p;
  float*       yph  = y + b * (CC * HWSZ) + 8 * half * HWSZ + pixw + p;
  const int*   labp = labels + qwave + p;
  const float* wl0  = &lw[p * CC + 2 * half];

#pragma unroll
  for (int t = 0; t < TILES_PER_WAVE; ++t) {
    const int lab = labp[t * 16];      // lanes 16-31 mirror lanes 0-15

    // x tile in B-matrix (4x16) chunk layout; offsets are immediates.
    float xv0[4], xv1[4];
#pragma unroll
    for (int j = 0; j < 4; ++j) {
      xv0[j] = xph[t * 16 + (4 * j) * HWSZ];
      xv1[j] = xph[t * 16 + (4 * j + 1) * HWSZ];
    }

    v8f acc = {};

    // Scalar-uniform scan over distinct labels present in this tile.
    unsigned long long rem = __ballot(1);
    while (rem) {
      const int src = (int)__builtin_ctzll(rem);
      const int l   = __shfl(lab, src, 32);      // broadcast -> uniform
      const bool m  = (lab == l);
      rem &= ~__ballot(m);

      const float* wl = wl0 + l * (CC * CC);     // one VALU add per label
#pragma unroll
      for (int j = 0; j < 4; ++j) {
        const v2f a = *(const v2f*)(wl + 4 * j); // ds_load_b64, imm offset
        v2f bv;                                  // mask non-matching columns
        bv.x = m ? xv0[j] : 0.0f;
        bv.y = m ? xv1[j] : 0.0f;
        acc = __builtin_amdgcn_wmma_f32_16x16x4_f32(
            /*neg_a=*/false, a, /*neg_b=*/false, bv,
            /*c_mod=*/(short)0, acc, /*reuse_a=*/false, /*reuse_b=*/false);
      }
    }

    // Reference adds bias[label] (zeros, but stay faithful).
    const float bp = bias[lab];

    // D layout: VGPR i -> row o = i + 8*half, column = p. Coalesced stores.
#pragma unroll
    for (int i = 0; i < 8; ++i)
      yph[t * 16 + i * HWSZ] = acc[i] + bp;
  }
}

extern "C" void kernel_launch(void* const* d_in, const int* in_sizes, int n_in,
                              void* d_out, int out_size, void* d_ws, size_t ws_size,
                              hipStream_t stream) {
  const float* x      = (const float*)d_in[0];
  const int*   labels = (const int*)d_in[1];
  const float* weight = (const float*)d_in[2];
  const float* bias   = (const float*)d_in[3];
  float*       y      = (float*)d_out;

  invnuconv_wmma_kernel<<<NBLOCKS, THREADS, 0, stream>>>(x, labels, weight, bias, y);
}